// RegionPooler_10282151707089
// MI455X (gfx1250) — compile-verified
//
#include <hip/hip_runtime.h>
#include <math.h>

#define B_  32
#define K_  64
#define N_  2304
#define D_  1024
#define H_  48
#define W_  48
#define EPSF 1e-6f
#define THRF 0.05f

typedef float v2f __attribute__((ext_vector_type(2)));
typedef float v8f __attribute__((ext_vector_type(8)));

// ---- CDNA5 async global->LDS copy (ASYNCcnt-tracked), with sync fallback ----
#if defined(__has_builtin)
#if __has_builtin(__builtin_amdgcn_global_load_async_to_lds_b128) && \
    __has_builtin(__builtin_amdgcn_s_wait_asynccnt)
#define USE_ASYNC 1
#endif
#endif
#ifndef USE_ASYNC
#define USE_ASYNC 0
#endif

// exact pointee type demanded by the builtin: 16-byte int vector
typedef int v4i __attribute__((vector_size(4 * sizeof(int))));
typedef v4i __attribute__((address_space(1))) gv4i;  // global
typedef v4i __attribute__((address_space(3))) lv4i;  // LDS

__device__ __forceinline__ void async_copy16(const float* g, float* l) {
#if USE_ASYNC
  __builtin_amdgcn_global_load_async_to_lds_b128((gv4i*)g, (lv4i*)l, 0, 0);
#else
  *(float4*)l = *(const float4*)g;
#endif
}

__device__ __forceinline__ void async_wait0() {
#if USE_ASYNC
  __builtin_amdgcn_s_wait_asynccnt(0);
#endif
}

// ---------------------------------------------------------------------------
// Kernel A: per-(b,k) reduction: geometry features + scores/sum normalizer
// ---------------------------------------------------------------------------
__global__ __launch_bounds__(256) void geom_stats_kernel(
    const float* __restrict__ mask,    // [B,K,N]
    const float* __restrict__ scores,  // [B,K]
    float* __restrict__ geom,          // [B*K,5]
    float* __restrict__ sinv)          // [B*K]
{
  __shared__ float s_sum[256];
  __shared__ int   s_cnt[256], s_ymin[256], s_ymax[256], s_xmin[256], s_xmax[256];

  const int bk  = blockIdx.x;
  const int tid = threadIdx.x;
  const float* row = mask + (size_t)bk * N_;

  float sum = 0.f;
  int cnt = 0, ymin = H_, ymax = -1, xmin = W_, xmax = -1;
  for (int i = tid; i < N_; i += 256) {
    float v = row[i];
    sum += fmaxf(v, 0.f);
    if (v > THRF) {
      int y = i / W_;
      int x = i - y * W_;
      cnt++;
      ymin = (y < ymin) ? y : ymin;
      ymax = (y > ymax) ? y : ymax;
      xmin = (x < xmin) ? x : xmin;
      xmax = (x > xmax) ? x : xmax;
    }
  }
  s_sum[tid] = sum; s_cnt[tid] = cnt;
  s_ymin[tid] = ymin; s_ymax[tid] = ymax;
  s_xmin[tid] = xmin; s_xmax[tid] = xmax;
  __syncthreads();
  for (int s = 128; s > 0; s >>= 1) {
    if (tid < s) {
      s_sum[tid] += s_sum[tid + s];
      s_cnt[tid] += s_cnt[tid + s];
      s_ymin[tid] = (s_ymin[tid + s] < s_ymin[tid]) ? s_ymin[tid + s] : s_ymin[tid];
      s_ymax[tid] = (s_ymax[tid + s] > s_ymax[tid]) ? s_ymax[tid + s] : s_ymax[tid];
      s_xmin[tid] = (s_xmin[tid + s] < s_xmin[tid]) ? s_xmin[tid + s] : s_xmin[tid];
      s_xmax[tid] = (s_xmax[tid + s] > s_xmax[tid]) ? s_xmax[tid + s] : s_xmax[tid];
    }
    __syncthreads();
  }
  if (tid == 0) {
    const float hd = (float)(H_ - 1), wd = (float)(W_ - 1);
    float ne = (s_cnt[0] > 0) ? 1.f : 0.f;
    geom[bk * 5 + 0] = ne * ((float)s_xmin[0] / wd);
    geom[bk * 5 + 1] = ne * ((float)s_ymin[0] / hd);
    geom[bk * 5 + 2] = ne * ((float)s_xmax[0] / wd);
    geom[bk * 5 + 3] = ne * ((float)s_ymax[0] / hd);
    geom[bk * 5 + 4] = ne * ((float)s_cnt[0] / (float)(H_ * W_));
    sinv[bk] = scores[bk] / fmaxf(s_sum[0], EPSF);
  }
}

// ---------------------------------------------------------------------------
// Kernel B: hidden = silu(geom @ W1 + b1)   [2048 x 1024]
// ---------------------------------------------------------------------------
__global__ __launch_bounds__(256) void mlp1_kernel(
    const float* __restrict__ geom,   // [B*K,5]
    const float* __restrict__ W1,     // [5,D]
    const float* __restrict__ b1,     // [D]
    float* __restrict__ hidden)       // [B*K,D]
{
  const int bk  = blockIdx.x;
  const int tid = threadIdx.x;
  const float g0 = geom[bk * 5 + 0];
  const float g1 = geom[bk * 5 + 1];
  const float g2 = geom[bk * 5 + 2];
  const float g3 = geom[bk * 5 + 3];
  const float g4 = geom[bk * 5 + 4];
#pragma unroll
  for (int i = 0; i < 4; ++i) {
    int d = tid + i * 256;
    float h = b1[d];
    h = fmaf(g0, W1[0 * D_ + d], h);
    h = fmaf(g1, W1[1 * D_ + d], h);
    h = fmaf(g2, W1[2 * D_ + d], h);
    h = fmaf(g3, W1[3 * D_ + d], h);
    h = fmaf(g4, W1[4 * D_ + d], h);
    hidden[(size_t)bk * D_ + d] = h / (1.f + expf(-h));   // silu
  }
}

// ---------------------------------------------------------------------------
// Kernel C: out = hidden @ W2 + b2 via V_WMMA_F32_16X16X4_F32
// grid (2048/16, 1024/128), block 256 (8 waves); k-chunk 32, double-buffered
// async global->LDS staging for both operands.
// ---------------------------------------------------------------------------
__global__ __launch_bounds__(256) void mlp2_wmma_kernel(
    const float* __restrict__ hidden, // [2048,1024]
    const float* __restrict__ W2,     // [1024,1024]
    const float* __restrict__ b2,     // [1024]
    float* __restrict__ out)          // [2048,1024]
{
  __shared__ float As[2][16][36];     // 16 rows x 32 k  (stride 36: 16B rows, no bank conflicts)
  __shared__ float Bs[2][32][144];    // 32 k x 128 cols (stride 144: half-waves on disjoint banks)

  const int tid  = threadIdx.x;
  const int wave = tid >> 5;
  const int lane = tid & 31;
  const int row0 = blockIdx.x * 16;
  const int nb   = blockIdx.y * 128;
  const int cB   = wave * 16 + (lane & 15);
  const int n    = nb + cB;
  const int kb   = (lane >> 4) * 2;   // K offset for this half-wave

  v8f c;
  const float bias = b2[n];
#pragma unroll
  for (int g = 0; g < 8; ++g) c[g] = bias;

  // staging: A tile 16x32 = 128 float4 (waves each copy 16; lanes 0..15 active)
  //          B tile 32x128 = 1024 float4 (4 per lane)
  auto stage = [&](int buf, int k0) {
    if (lane < 16) {
      int idx = wave * 16 + lane;       // 0..127
      int r = idx >> 3, c4 = idx & 7;   // 8 float4 per A row
      async_copy16(hidden + (size_t)(row0 + r) * D_ + k0 + c4 * 4, &As[buf][r][c4 * 4]);
    }
#pragma unroll
    for (int i = 0; i < 4; ++i) {
      int idx = wave * 128 + i * 32 + lane;  // 0..1023
      int r = idx >> 5, c4 = idx & 31;       // 32 float4 per B row
      async_copy16(W2 + (size_t)(k0 + r) * D_ + nb + c4 * 4, &Bs[buf][r][c4 * 4]);
    }
  };

  stage(0, 0);
  int buf = 0;
  for (int k0 = 0; k0 < D_; k0 += 32) {
    async_wait0();
    __syncthreads();                       // tile ready; prev compute done
    if (k0 + 32 < D_) stage(buf ^ 1, k0 + 32);
#pragma unroll
    for (int kk = 0; kk < 32; kk += 4) {
      v2f a;
      a.x = As[buf][lane & 15][kk + kb];
      a.y = As[buf][lane & 15][kk + kb + 1];
      v2f bf;
      bf.x = Bs[buf][kk + kb][cB];
      bf.y = Bs[buf][kk + kb + 1][cB];
      c = __builtin_amdgcn_wmma_f32_16x16x4_f32(false, a, false, bf,
                                                (short)0, c, false, false);
    }
    buf ^= 1;
  }

  const int mbase = (lane >> 4) * 8;
#pragma unroll
  for (int g = 0; g < 8; ++g)
    out[(size_t)(row0 + mbase + g) * D_ + n] = c[g];
}

// ---------------------------------------------------------------------------
// Kernel D: out += (scores*relu(mask)/sum) @ patch_tokens  (per batch)
// grid (B, 1024/128), block 256 (8 waves); k-chunk 32, double-buffered async
// staging; relu*scale applied at fragment build (co-executes with XDL WMMA).
// ---------------------------------------------------------------------------
__global__ __launch_bounds__(256) void pool_wmma_kernel(
    const float* __restrict__ patch,  // [B,N,D]
    const float* __restrict__ mask,   // [B,K,N]
    const float* __restrict__ sinv,   // [B*K]
    float* __restrict__ out)          // [B,K,D]
{
  __shared__ float As[2][64][36];     // 64 regions x 32 k (raw mask)
  __shared__ float Bs[2][32][144];    // 32 k x 128 cols of patch

  const int tid  = threadIdx.x;
  const int wave = tid >> 5;
  const int lane = tid & 31;
  const int b    = blockIdx.x;
  const int nb   = blockIdx.y * 128;
  const int cB   = wave * 16 + (lane & 15);
  const int n    = nb + cB;
  const int kb   = (lane >> 4) * 2;

  const float* P  = patch + (size_t)b * N_ * D_;
  const float* Mb = mask  + (size_t)b * K_ * N_;
  const size_t obase = (size_t)b * K_ * D_ + n;

  // per-lane A-row scales (score/sum for the 4 region rows this lane touches)
  float scm[4];
#pragma unroll
  for (int mt = 0; mt < 4; ++mt)
    scm[mt] = sinv[b * K_ + mt * 16 + (lane & 15)];

  v8f acc[4];
  const int mbase = (lane >> 4) * 8;
#pragma unroll
  for (int mt = 0; mt < 4; ++mt)
#pragma unroll
    for (int g = 0; g < 8; ++g)
      acc[mt][g] = out[obase + (size_t)(mt * 16 + mbase + g) * D_];

  // staging: A tile 64x32 = 512 float4 (2 per lane); B tile 32x128 = 1024 f4 (4/lane)
  auto stage = [&](int buf, int j0) {
#pragma unroll
    for (int i = 0; i < 2; ++i) {
      int idx = wave * 64 + i * 32 + lane;  // 0..511
      int r = idx >> 3, c4 = idx & 7;       // 8 float4 per A row
      async_copy16(Mb + (size_t)r * N_ + j0 + c4 * 4, &As[buf][r][c4 * 4]);
    }
#pragma unroll
    for (int i = 0; i < 4; ++i) {
      int idx = wave * 128 + i * 32 + lane; // 0..1023
      int r = idx >> 5, c4 = idx & 31;      // 32 float4 per B row
      async_copy16(P + (size_t)(j0 + r) * D_ + nb + c4 * 4, &Bs[buf][r][c4 * 4]);
    }
  };

  stage(0, 0);
  int buf = 0;
  for (int j0 = 0; j0 < N_; j0 += 32) {
    async_wait0();
    __syncthreads();
    if (j0 + 32 < N_) stage(buf ^ 1, j0 + 32);
#pragma unroll
    for (int kk = 0; kk < 32; kk += 4) {
      v2f bf;
      bf.x = Bs[buf][kk + kb][cB];
      bf.y = Bs[buf][kk + kb + 1][cB];
#pragma unroll
      for (int mt = 0; mt < 4; ++mt) {
        float a0 = As[buf][mt * 16 + (lane & 15)][kk + kb];
        float a1 = As[buf][mt * 16 + (lane & 15)][kk + kb + 1];
        v2f a;
        a.x = fmaxf(a0, 0.f) * scm[mt];
        a.y = fmaxf(a1, 0.f) * scm[mt];
        acc[mt] = __builtin_amdgcn_wmma_f32_16x16x4_f32(false, a, false, bf,
                                                        (short)0, acc[mt],
                                                        false, false);
      }
    }
    buf ^= 1;
  }

#pragma unroll
  for (int mt = 0; mt < 4; ++mt)
#pragma unroll
    for (int g = 0; g < 8; ++g)
      out[obase + (size_t)(mt * 16 + mbase + g) * D_] = acc[mt][g];
}

// ---------------------------------------------------------------------------
extern "C" void kernel_launch(void* const* d_in, const int* in_sizes, int n_in,
                              void* d_out, int out_size, void* d_ws, size_t ws_size,
                              hipStream_t stream) {
  const float* patch  = (const float*)d_in[0];  // [B,N,D]
  const float* mask   = (const float*)d_in[1];  // [B,K,N]
  const float* scores = (const float*)d_in[2];  // [B,K]
  const float* W1     = (const float*)d_in[3];  // [5,D]
  const float* b1     = (const float*)d_in[4];  // [D]
  const float* W2     = (const float*)d_in[5];  // [D,D]
  const float* b2     = (const float*)d_in[6];  // [D]
  float* out = (float*)d_out;                   // [B,K,D]

  char* ws = (char*)d_ws;
  float* geom   = (float*)ws;                          // 2048*5 floats
  float* sinv   = (float*)(ws + (size_t)2048 * 5 * 4); // 2048 floats
  float* hidden = (float*)(ws + (size_t)2048 * 6 * 4); // 2048*1024 floats (8 MB)

  geom_stats_kernel<<<B_ * K_, 256, 0, stream>>>(mask, scores, geom, sinv);
  mlp1_kernel<<<B_ * K_, 256, 0, stream>>>(geom, W1, b1, hidden);
  mlp2_wmma_kernel<<<dim3((B_ * K_) / 16, D_ / 128), 256, 0, stream>>>(hidden, W2, b2, out);
  pool_wmma_kernel<<<dim3(B_, D_ / 128), 256, 0, stream>>>(patch, mask, sinv, out);
}